// GraphEmbedder_32349693673917
// MI455X (gfx1250) — compile-verified
//
#include <hip/hip_runtime.h>

typedef __attribute__((ext_vector_type(2))) float v2f;
typedef __attribute__((ext_vector_type(8))) float v8f;

#define N_NODES_C 100000
#define N_GRAPHS_C 512
#define IN_CH 21
#define HID 64

// workspace layout (float offsets)
constexpr size_t OFF_DEG  = 0;                                     // [N]
constexpr size_t OFF_AGG1 = OFF_DEG  + (size_t)N_NODES_C;          // [N*21]
constexpr size_t OFF_H    = OFF_AGG1 + (size_t)N_NODES_C * IN_CH;  // [N*64]
constexpr size_t OFF_AGG2 = OFF_H    + (size_t)N_NODES_C * HID;    // [N*64]
constexpr size_t OFF_GSUM = OFF_AGG2 + (size_t)N_NODES_C * HID;    // [512*64]
constexpr size_t OFF_GCNT = OFF_GSUM + (size_t)N_GRAPHS_C * HID;   // [512]
constexpr size_t WS_FLOATS = OFF_GCNT + (size_t)N_GRAPHS_C;

__global__ void k_zero(float* __restrict__ p, long n) {
    long i = (long)blockIdx.x * blockDim.x + threadIdx.x;
    long stride = (long)gridDim.x * blockDim.x;
    for (; i < n; i += stride) p[i] = 0.0f;
}

__global__ void k_degree(const int* __restrict__ dst, float* __restrict__ deg, int nE) {
    int e = blockIdx.x * blockDim.x + threadIdx.x;
    if (e < nE) atomicAdd(&deg[dst[e]], 1.0f);
}

__global__ void k_gcnt(const int* __restrict__ batch, float* __restrict__ gcnt, int nN) {
    int n = blockIdx.x * blockDim.x + threadIdx.x;
    if (n < nN) atomicAdd(&gcnt[batch[n]], 1.0f);
}

// layer-1 message scatter: one thread per (edge, channel), 21 channels
__global__ void k_scatter1(const float* __restrict__ x, const int* __restrict__ src,
                           const int* __restrict__ dst, float* __restrict__ agg1, int nE) {
    long t = (long)blockIdx.x * blockDim.x + threadIdx.x;
    long total = (long)nE * IN_CH;
    if (t >= total) return;
    int e = (int)(t / IN_CH);
    int c = (int)(t - (long)e * IN_CH);
    float v = x[(long)src[e] * IN_CH + c];
    atomicAdd(&agg1[(long)dst[e] * IN_CH + c], v);
}

// layer-2 message scatter: float4 per thread, 16 threads per edge (64 ch)
__global__ void k_scatter2(const float* __restrict__ h, const int* __restrict__ src,
                           const int* __restrict__ dst, float* __restrict__ agg2, int nE) {
    long t = (long)blockIdx.x * blockDim.x + threadIdx.x;
    long total = (long)nE * 16;
    if (t >= total) return;
    int e = (int)(t >> 4);
    int c = (int)(t & 15) * 4;
    const float4 v = *reinterpret_cast<const float4*>(h + (long)src[e] * HID + c);
    float* o = agg2 + (long)dst[e] * HID + c;
    atomicAdd(o + 0, v.x);
    atomicAdd(o + 1, v.y);
    atomicAdd(o + 2, v.z);
    atomicAdd(o + 3, v.w);
}

// SAGE layer 1: h = relu( (agg1/deg) @ W1l + b1 + x @ W1r ), via fp32 WMMA.
// One wave handles 16 nodes x 64 output channels. K = 21 padded to 24.
__global__ void __launch_bounds__(256) k_sage1(
    const float* __restrict__ x, const float* __restrict__ agg1,
    const float* __restrict__ deg,
    const float* __restrict__ Wl, const float* __restrict__ bias,
    const float* __restrict__ Wr, float* __restrict__ h, int nN)
{
    const int lane = threadIdx.x & 31;
    const int wave = threadIdx.x >> 5;
    const int base = (blockIdx.x * 8 + wave) * 16;
    if (base >= nN) return;                 // wave-uniform; EXEC all-1s below

    const int hf  = lane >> 4;              // 0: K pair {0,1}, 1: K pair {2,3}
    const int l16 = lane & 15;
    const int arow = base + l16;            // A-matrix row for this lane
    const float inv = 1.0f / fmaxf(deg[arow], 1.0f);

    v8f accL[4] = {};
    v8f accR[4] = {};

    #pragma unroll
    for (int k0 = 0; k0 < 24; k0 += 4) {
        const int c0 = k0 + hf * 2;
        const int c1 = c0 + 1;
        v2f aA, aX;
        aA.x = (c0 < IN_CH) ? agg1[(long)arow * IN_CH + c0] * inv : 0.0f;
        aA.y = (c1 < IN_CH) ? agg1[(long)arow * IN_CH + c1] * inv : 0.0f;
        aX.x = (c0 < IN_CH) ? x[(long)arow * IN_CH + c0] : 0.0f;
        aX.y = (c1 < IN_CH) ? x[(long)arow * IN_CH + c1] : 0.0f;
        #pragma unroll
        for (int j = 0; j < 4; ++j) {
            const int col = j * 16 + l16;
            v2f bL, bR;
            bL.x = (c0 < IN_CH) ? Wl[c0 * HID + col] : 0.0f;
            bL.y = (c1 < IN_CH) ? Wl[c1 * HID + col] : 0.0f;
            bR.x = (c0 < IN_CH) ? Wr[c0 * HID + col] : 0.0f;
            bR.y = (c1 < IN_CH) ? Wr[c1 * HID + col] : 0.0f;
            accL[j] = __builtin_amdgcn_wmma_f32_16x16x4_f32(
                false, aA, false, bL, (short)0, accL[j], false, false);
            accR[j] = __builtin_amdgcn_wmma_f32_16x16x4_f32(
                false, aX, false, bR, (short)0, accR[j], false, false);
        }
    }

    #pragma unroll
    for (int j = 0; j < 4; ++j) {
        const int col = j * 16 + l16;
        const float b = bias[col];
        #pragma unroll
        for (int r = 0; r < 8; ++r) {
            const int row = base + hf * 8 + r;   // D layout: lanes 16-31 hold M=8..15
            float v = accL[j][r] + accR[j][r] + b;
            h[(long)row * HID + col] = fmaxf(v, 0.0f);
        }
    }
}

// SAGE layer 2 + fused global mean-pool numerator:
// h2 = (agg2/deg) @ W2l + b2 + h @ W2r ; atomicAdd into gsum[batch[node]]
__global__ void __launch_bounds__(256) k_sage2(
    const float* __restrict__ h, const float* __restrict__ agg2,
    const float* __restrict__ deg,
    const float* __restrict__ Wl, const float* __restrict__ bias,
    const float* __restrict__ Wr, const int* __restrict__ batch,
    float* __restrict__ gsum, int nN)
{
    const int lane = threadIdx.x & 31;
    const int wave = threadIdx.x >> 5;
    const int base = (blockIdx.x * 8 + wave) * 16;
    if (base >= nN) return;                 // wave-uniform

    const int hf  = lane >> 4;
    const int l16 = lane & 15;
    const int arow = base + l16;
    const float inv = 1.0f / fmaxf(deg[arow], 1.0f);

    v8f accL[4] = {};
    v8f accR[4] = {};

    for (int k0 = 0; k0 < HID; k0 += 4) {
        const int c0 = k0 + hf * 2;
        const int c1 = c0 + 1;
        v2f aA, aX;
        aA.x = agg2[(long)arow * HID + c0] * inv;
        aA.y = agg2[(long)arow * HID + c1] * inv;
        aX.x = h[(long)arow * HID + c0];
        aX.y = h[(long)arow * HID + c1];
        #pragma unroll
        for (int j = 0; j < 4; ++j) {
            const int col = j * 16 + l16;
            v2f bL, bR;
            bL.x = Wl[c0 * HID + col];
            bL.y = Wl[c1 * HID + col];
            bR.x = Wr[c0 * HID + col];
            bR.y = Wr[c1 * HID + col];
            accL[j] = __builtin_amdgcn_wmma_f32_16x16x4_f32(
                false, aA, false, bL, (short)0, accL[j], false, false);
            accR[j] = __builtin_amdgcn_wmma_f32_16x16x4_f32(
                false, aX, false, bR, (short)0, accR[j], false, false);
        }
    }

    #pragma unroll
    for (int j = 0; j < 4; ++j) {
        const int col = j * 16 + l16;
        const float b = bias[col];
        #pragma unroll
        for (int r = 0; r < 8; ++r) {
            const int row = base + hf * 8 + r;
            const int g = batch[row];
            float v = accL[j][r] + accR[j][r] + b;
            atomicAdd(&gsum[(long)g * HID + col], v);
        }
    }
}

__global__ void k_final(const float* __restrict__ gsum, const float* __restrict__ gcnt,
                        float* __restrict__ out) {
    int i = blockIdx.x * blockDim.x + threadIdx.x;
    if (i < N_GRAPHS_C * HID) {
        int g = i / HID;
        out[i] = gsum[i] / fmaxf(gcnt[g], 1.0f);
    }
}

extern "C" void kernel_launch(void* const* d_in, const int* in_sizes, int n_in,
                              void* d_out, int out_size, void* d_ws, size_t ws_size,
                              hipStream_t stream) {
    (void)n_in; (void)out_size; (void)ws_size;
    const float* x    = (const float*)d_in[0];
    const int*   ei   = (const int*)d_in[1];
    const int*   batch= (const int*)d_in[2];
    const float* W1l  = (const float*)d_in[3];
    const float* b1   = (const float*)d_in[4];
    const float* W1r  = (const float*)d_in[5];
    const float* W2l  = (const float*)d_in[6];
    const float* b2   = (const float*)d_in[7];
    const float* W2r  = (const float*)d_in[8];
    float* out = (float*)d_out;
    float* ws  = (float*)d_ws;

    const int nE = in_sizes[1] / 2;
    const int nN = in_sizes[0] / IN_CH;
    const int* src = ei;
    const int* dst = ei + nE;

    float* deg  = ws + OFF_DEG;
    float* agg1 = ws + OFF_AGG1;
    float* h    = ws + OFF_H;
    float* agg2 = ws + OFF_AGG2;
    float* gsum = ws + OFF_GSUM;
    float* gcnt = ws + OFF_GCNT;

    // zero deg+agg1 and agg2+gsum+gcnt (h is fully overwritten by k_sage1)
    k_zero<<<2048, 256, 0, stream>>>(deg,  (long)(OFF_H - OFF_DEG));
    k_zero<<<2048, 256, 0, stream>>>(agg2, (long)(WS_FLOATS - OFF_AGG2));

    k_degree<<<(nE + 255) / 256, 256, 0, stream>>>(dst, deg, nE);
    k_gcnt<<<(nN + 255) / 256, 256, 0, stream>>>(batch, gcnt, nN);

    long t1 = (long)nE * IN_CH;
    k_scatter1<<<(int)((t1 + 255) / 256), 256, 0, stream>>>(x, src, dst, agg1, nE);

    int ntiles = (nN + 15) / 16;
    int nblk   = (ntiles + 7) / 8;
    k_sage1<<<nblk, 256, 0, stream>>>(x, agg1, deg, W1l, b1, W1r, h, nN);

    long t2 = (long)nE * 16;
    k_scatter2<<<(int)((t2 + 255) / 256), 256, 0, stream>>>(h, src, dst, agg2, nE);

    k_sage2<<<nblk, 256, 0, stream>>>(h, agg2, deg, W2l, b2, W2r, batch, gsum, nN);

    k_final<<<(N_GRAPHS_C * HID + 255) / 256, 256, 0, stream>>>(gsum, gcnt, out);
}